// LoRAFALiMELinear_17325898072230
// MI455X (gfx1250) — compile-verified
//
#include <hip/hip_runtime.h>
#include <hip/hip_bf16.h>
#include <math.h>

typedef __attribute__((ext_vector_type(2))) float v2f;
typedef __attribute__((ext_vector_type(8))) float v8f;

#define D_IN   4096
#define D_OUT  4096
#define RANK_  32
#define NEXP   8
#define TOKENS 8192      // Bsz*T = 4*2048
#define ANCH   4096      // Bsz * T/2
#define SCALING 0.5f     // ALPHA/RANK
#define GAMMA_ 0.5f
#define STT    0.5f      // SOFT_TOPK_TEMP
#define EPS_   1e-6f

// ---------------- kernel 1: z = x @ A  (8192x4096 @ 4096x32) ----------------
__global__ void k_z(const float* __restrict__ x, const float* __restrict__ A,
                    float* __restrict__ z) {
    int r   = threadIdx.x & 31;
    int row = blockIdx.x * 8 + (threadIdx.x >> 5);
    const float4* x4 = (const float4*)(x + (size_t)row * D_IN);
    float acc = 0.f;
#pragma unroll 4
    for (int d4 = 0; d4 < D_IN / 4; ++d4) {
        float4 v = x4[d4];
        int d = d4 * 4;
        acc = fmaf(v.x, A[(size_t)(d + 0) * RANK_ + r], acc);
        acc = fmaf(v.y, A[(size_t)(d + 1) * RANK_ + r], acc);
        acc = fmaf(v.z, A[(size_t)(d + 2) * RANK_ + r], acc);
        acc = fmaf(v.w, A[(size_t)(d + 3) * RANK_ + r], acc);
    }
    z[(size_t)row * RANK_ + r] = acc;
}

// ------- kernel 2: H_slice[ar,e] = x[anchor] . W[e] + b[e]; max|.| --------
__global__ void k_hslice(const float* __restrict__ x, const float* __restrict__ W,
                         const float* __restrict__ bvec, float* __restrict__ Hs,
                         unsigned* __restrict__ scl) {
    int e  = threadIdx.x & 7;
    int ar = blockIdx.x * 32 + (threadIdx.x >> 3);     // 0..4095
    int bi = ar >> 10, a = ar & 1023;
    size_t row = (size_t)bi * 2048 + 2 * a + 1;        // anchor token
    const float4* x4 = (const float4*)(x + row * D_IN);
    const float4* W4 = (const float4*)(W + (size_t)e * D_IN);
    float acc = bvec[e];
#pragma unroll 4
    for (int d = 0; d < D_IN / 4; ++d) {
        float4 xa = x4[d], wa = W4[d];
        acc = fmaf(xa.x, wa.x, acc);
        acc = fmaf(xa.y, wa.y, acc);
        acc = fmaf(xa.z, wa.z, acc);
        acc = fmaf(xa.w, wa.w, acc);
    }
    Hs[(size_t)ar * NEXP + e] = acc;
    atomicMax(scl, __float_as_uint(fabsf(acc)));       // abs-float bits order as uint
}

// ------- kernel 3: delta_slice[ar,e] = z[anchor] . LiMEs[e]; max|.| -------
__global__ void k_dslice(const float* __restrict__ z, const float* __restrict__ L,
                         float* __restrict__ Ds, unsigned* __restrict__ scl) {
    int e  = threadIdx.x & 7;
    int ar = blockIdx.x * 32 + (threadIdx.x >> 3);
    int bi = ar >> 10, a = ar & 1023;
    size_t row = (size_t)bi * 2048 + 2 * a + 1;
    const float* zr = z + row * RANK_;
    const float* Lr = L + e * RANK_;
    float acc = 0.f;
#pragma unroll
    for (int r = 0; r < RANK_; ++r) acc = fmaf(zr[r], Lr[r], acc);
    Ds[(size_t)ar * NEXP + e] = acc;
    atomicMax(scl, __float_as_uint(fabsf(acc)));
}

// ---- kernel 4: per-anchor softmax / soft-top2 / weights; zp = z*pmix*s ----
__global__ void k_mix(const float* __restrict__ Hs, const float* __restrict__ Ds,
                      const float* __restrict__ L, const unsigned* __restrict__ scl,
                      const float* __restrict__ z, float* __restrict__ zp) {
    int ar = blockIdx.x * blockDim.x + threadIdx.x;    // 0..4095
    float hsc = fmaxf(__uint_as_float(scl[0]), EPS_);
    float dsc = fmaxf(__uint_as_float(scl[1]), EPS_);
    float p[NEXP];
    float mx = -1e30f;
#pragma unroll
    for (int e = 0; e < NEXP; ++e) {
        float lg = (1.0f - GAMMA_) * (Hs[(size_t)ar * NEXP + e] / hsc)
                 + GAMMA_          * (Ds[(size_t)ar * NEXP + e] / dsc);
        p[e] = lg;                          // TEMPERATURE = 1
        mx = fmaxf(mx, lg);
    }
    float s = 0.f;
#pragma unroll
    for (int e = 0; e < NEXP; ++e) { p[e] = __expf(p[e] - mx); s += p[e]; }
    float inv = 1.0f / s;
#pragma unroll
    for (int e = 0; e < NEXP; ++e) p[e] *= inv;
    float m1 = -1e30f, m2 = -1e30f;         // top-2 of probs
#pragma unroll
    for (int e = 0; e < NEXP; ++e) {
        float v = p[e];
        if (v > m1) { m2 = m1; m1 = v; } else m2 = fmaxf(m2, v);
    }
    float w[NEXP]; float sw = 0.f;
#pragma unroll
    for (int e = 0; e < NEXP; ++e) {
        float mask = 1.0f / (1.0f + __expf(-(p[e] - m2) / STT));
        w[e] = p[e] * mask; sw += w[e];
    }
    float inv2 = 1.0f / (sw + 1e-9f);
#pragma unroll
    for (int e = 0; e < NEXP; ++e) w[e] *= inv2;

    int bi = ar >> 10, a = ar & 1023;
    size_t t0 = (size_t)bi * 2048 + 2 * a;             // this anchor covers t0, t0+1
#pragma unroll
    for (int r = 0; r < RANK_; ++r) {
        float pm = 0.f;
#pragma unroll
        for (int e = 0; e < NEXP; ++e) pm = fmaf(w[e], L[e * RANK_ + r], pm);
        pm *= SCALING;
        zp[t0 * RANK_ + r]       = z[t0 * RANK_ + r]       * pm;
        zp[(t0 + 1) * RANK_ + r] = z[(t0 + 1) * RANK_ + r] * pm;
    }
}

// ---------------- kernel 5: out = x@Wt + zp@Bmat + b  (WMMA f32) ----------------
#define BM 128
#define BN 128
#define BK 16
#define LDP 20   // LDS pitch: rows 16B-aligned, conflict-free frag reads

#if defined(__HIP_DEVICE_COMPILE__) && \
    __has_builtin(__builtin_amdgcn_global_load_async_to_lds_b128) && \
    __has_builtin(__builtin_amdgcn_s_wait_asynccnt)
#define USE_ASYNC_LDS 1
// builtin signature (from clang diagnostic): param0 = global-AS int4*, so cast exactly.
typedef int v4i_t __attribute__((vector_size(16)));
typedef __attribute__((address_space(1))) v4i_t glob_v4i;
typedef __attribute__((address_space(3))) v4i_t lds_v4i;
__device__ __forceinline__ void cp_async16(const float* g, float* l) {
    __builtin_amdgcn_global_load_async_to_lds_b128(
        (glob_v4i*)g, (lds_v4i*)l, 0, 0);
}
#else
#define USE_ASYNC_LDS 0
#endif

// one BK chunk: 4 k-steps x (2 M-tiles x 4 N-tiles) = 32 wmma per wave
__device__ __forceinline__ void wmma_step(const float* __restrict__ as,
                                          const float* __restrict__ bs,
                                          int wm, int wn, int lm, int lk,
                                          v8f (&acc)[2][4]) {
#pragma unroll
    for (int ks = 0; ks < BK; ks += 4) {
        v2f af[2], bf[4];
#pragma unroll
        for (int i = 0; i < 2; ++i) {
            const float* p = &as[(wm + i * 16 + lm) * LDP + ks + lk];
            af[i][0] = p[0]; af[i][1] = p[1];
        }
#pragma unroll
        for (int j = 0; j < 4; ++j) {
            const float* p = &bs[(wn + j * 16 + lm) * LDP + ks + lk];
            bf[j][0] = p[0]; bf[j][1] = p[1];
        }
#pragma unroll
        for (int i = 0; i < 2; ++i)
#pragma unroll
            for (int j = 0; j < 4; ++j)
                acc[i][j] = __builtin_amdgcn_wmma_f32_16x16x4_f32(
                    false, af[i], false, bf[j], (short)0, acc[i][j], false, false);
    }
}

__global__ __launch_bounds__(256)
void k_gemm(const float* __restrict__ x, const float* __restrict__ W,
            const float* __restrict__ bvec, const float* __restrict__ zp,
            const float* __restrict__ Bmat, float* __restrict__ out) {
#if USE_ASYNC_LDS
    __shared__ float As[3][BM * LDP];
    __shared__ float Bs[3][BN * LDP];
#else
    __shared__ float As[2][BM * LDP];
    __shared__ float Bs[2][BN * LDP];
#endif
    const int tid  = threadIdx.x;
    const int lane = tid & 31;
    const int wave = tid >> 5;
    const int wm   = (wave >> 1) * 32;
    const int wn   = (wave & 1) * 64;
    const int tM   = blockIdx.y * BM;
    const int tN   = blockIdx.x * BN;
    const int lm   = lane & 15;
    const int lk   = (lane >> 4) * 2;

    // tile-load coordinates (identical pattern for A and B tiles)
    const int r0 = tid >> 2;            // 0..63
    const int c0 = (tid & 3) << 2;      // 0,4,8,12
    const float* gA = x + (size_t)(tM + r0) * D_IN + c0;
    const float* gB = W + (size_t)(tN + r0) * D_IN + c0;
    const size_t rstep = (size_t)64 * D_IN;

    v8f acc[2][4];
#pragma unroll
    for (int i = 0; i < 2; ++i)
#pragma unroll
        for (int j = 0; j < 4; ++j)
#pragma unroll
            for (int v = 0; v < 8; ++v) acc[i][j][v] = 0.f;

    const int NIT = D_IN / BK;          // 256

#if USE_ASYNC_LDS
    // ---- 3-buffer async pipeline: loads run 2 tiles ahead, 1 barrier/iter ----
    auto issue = [&](int b, int k0) {
        cp_async16(gA + k0,         &As[b][r0 * LDP + c0]);
        cp_async16(gA + rstep + k0, &As[b][(r0 + 64) * LDP + c0]);
        cp_async16(gB + k0,         &Bs[b][r0 * LDP + c0]);
        cp_async16(gB + rstep + k0, &Bs[b][(r0 + 64) * LDP + c0]);
    };
    issue(0, 0);
    issue(1, BK);
    for (int i = 0; i < NIT; ++i) {
        if (i + 1 < NIT) __builtin_amdgcn_s_wait_asynccnt(4);  // tile i landed
        else             __builtin_amdgcn_s_wait_asynccnt(0);
        __syncthreads();
        if (i + 2 < NIT) issue((i + 2) % 3, (i + 2) * BK);
        wmma_step(As[i % 3], Bs[i % 3], wm, wn, lm, lk, acc);
    }
#else
    // ---- register-staged 2-buffer pipeline: 1 barrier/iter ----
    float4 ra0, ra1, rb0, rb1;
    auto gload = [&](int k0) {
        ra0 = *(const float4*)(gA + k0);
        ra1 = *(const float4*)(gA + rstep + k0);
        rb0 = *(const float4*)(gB + k0);
        rb1 = *(const float4*)(gB + rstep + k0);
    };
    auto lstore = [&](int b) {
        *(float4*)&As[b][r0 * LDP + c0]        = ra0;
        *(float4*)&As[b][(r0 + 64) * LDP + c0] = ra1;
        *(float4*)&Bs[b][r0 * LDP + c0]        = rb0;
        *(float4*)&Bs[b][(r0 + 64) * LDP + c0] = rb1;
    };
    gload(0); lstore(0);
    for (int i = 0; i < NIT; ++i) {
        if (i + 1 < NIT) gload((i + 1) * BK);
        __syncthreads();
        wmma_step(As[i & 1], Bs[i & 1], wm, wn, lm, lk, acc);
        if (i + 1 < NIT) lstore((i + 1) & 1);
    }
#endif

    // ---- K extension: A = zp (row len 32), B = Bmat (k-major -> transpose) ----
    for (int k0 = 0; k0 < RANK_; k0 += BK) {
        __syncthreads();
        {
            float4 va = *(const float4*)(zp + (size_t)(tM + r0) * RANK_ + k0 + c0);
            *(float4*)&As[0][r0 * LDP + c0] = va;
            float4 vb = *(const float4*)(zp + (size_t)(tM + r0 + 64) * RANK_ + k0 + c0);
            *(float4*)&As[0][(r0 + 64) * LDP + c0] = vb;
        }
#pragma unroll
        for (int it = 0; it < 8; ++it) {        // 16x128 scalars, coalesced in n
            int idx = tid + it * 256;
            int k = idx >> 7, n = idx & 127;
            Bs[0][n * LDP + k] = Bmat[(size_t)(k0 + k) * D_OUT + tN + n];
        }
        __syncthreads();
        wmma_step(As[0], Bs[0], wm, wn, lm, lk, acc);
    }

    // ---- epilogue: + bias, store.  C layout: m = v + 8*(lane>>4), n = lane&15 ----
    const int hi8 = (lane >> 4) * 8;
#pragma unroll
    for (int i = 0; i < 2; ++i)
#pragma unroll
        for (int j = 0; j < 4; ++j) {
            int gn = tN + wn + j * 16 + lm;
            float bb = bvec[gn];
            int mb = tM + wm + i * 16 + hi8;
#pragma unroll
            for (int v = 0; v < 8; ++v)
                out[(size_t)(mb + v) * D_OUT + gn] = acc[i][j][v] + bb;
        }
}

// ------------------------------- launcher -------------------------------
extern "C" void kernel_launch(void* const* d_in, const int* in_sizes, int n_in,
                              void* d_out, int out_size, void* d_ws, size_t ws_size,
                              hipStream_t stream) {
    const float* x  = (const float*)d_in[0];
    const float* W  = (const float*)d_in[1];
    const float* b  = (const float*)d_in[2];
    const float* A  = (const float*)d_in[3];
    const float* Bm = (const float*)d_in[4];
    const float* L  = (const float*)d_in[5];
    float* out = (float*)d_out;

    float* ws  = (float*)d_ws;
    float* z   = ws;                                   // 8192*32
    float* zp  = z  + (size_t)TOKENS * RANK_;          // 8192*32
    float* Hs  = zp + (size_t)TOKENS * RANK_;          // 4096*8
    float* Ds  = Hs + (size_t)ANCH * NEXP;             // 4096*8
    unsigned* scl = (unsigned*)(Ds + (size_t)ANCH * NEXP);  // 2 uints

    (void)hipMemsetAsync(scl, 0, 2 * sizeof(unsigned), stream);

    k_z     <<<TOKENS / 8, 256, 0, stream>>>(x, A, z);
    k_hslice<<<ANCH / 32, 256, 0, stream>>>(x, W, b, Hs, scl);
    k_dslice<<<ANCH / 32, 256, 0, stream>>>(z, L, Ds, scl + 1);
    k_mix   <<<ANCH / 128, 128, 0, stream>>>(Hs, Ds, L, scl, z, zp);

    dim3 grid(D_OUT / BN, TOKENS / BM);   // (32, 64)
    k_gemm<<<grid, 256, 0, stream>>>(x, W, b, zp, Bm, out);
}